// graph_convolution_layer_5471788335182
// MI455X (gfx1250) — compile-verified
//
#include <hip/hip_runtime.h>

typedef float v2f __attribute__((ext_vector_type(2)));
typedef float v8f __attribute__((ext_vector_type(8)));

#define IN_F   256
#define HID_F  256
#define OUT_F  128
#define OUT_STRIDE 512   // concat(self_out[128], h1[256], h2[128])

// ---------------------------------------------------------------------------
// Guaranteed HW fp32 scatter-add: global_atomic_add_f32 (non-returning).
// ---------------------------------------------------------------------------
__device__ __forceinline__ void atomic_add_f32(float* p, float v) {
    asm volatile("global_atomic_add_f32 %0, %1, off"
                 :: "v"(p), "v"(v) : "memory");
}

// ---------------------------------------------------------------------------
// SpMM: agg[dst] += w * x[src].  One wave per edge; each lane moves 8 floats
// (2x float4 gather), 8 fp32 scatter-atomics.  x rows are 256 floats with
// arbitrary row stride ldx (512 when reading h1 out of the output buffer).
// ---------------------------------------------------------------------------
__global__ void spmm_kernel(const float* __restrict__ x, int ldx,
                            const int* __restrict__ esrc,
                            const int* __restrict__ edst,
                            const float* __restrict__ ew,
                            float* __restrict__ agg,   // [N, 256] accumulators
                            int E)
{
    const int lane = threadIdx.x & 31;
    const int wid  = (blockIdx.x * blockDim.x + threadIdx.x) >> 5;
    const int nw   = (gridDim.x * blockDim.x) >> 5;

    for (int e = wid; e < E; e += nw) {
        const int   s = esrc[e];
        const int   d = edst[e];
        const float w = ew[e];
        const float4* row = (const float4*)(x + (size_t)s * ldx);
        const float4 a = row[lane];        // floats [lane*4 .. lane*4+3]
        const float4 b = row[lane + 32];   // floats [128+lane*4 ..]
        float* o = agg + (size_t)d * HID_F;
        atomic_add_f32(o + lane * 4 + 0,       w * a.x);
        atomic_add_f32(o + lane * 4 + 1,       w * a.y);
        atomic_add_f32(o + lane * 4 + 2,       w * a.z);
        atomic_add_f32(o + lane * 4 + 3,       w * a.w);
        atomic_add_f32(o + 128 + lane * 4 + 0, w * b.x);
        atomic_add_f32(o + 128 + lane * 4 + 1, w * b.y);
        atomic_add_f32(o + 128 + lane * 4 + 2, w * b.z);
        atomic_add_f32(o + 128 + lane * 4 + 3, w * b.w);
    }
}

// ---------------------------------------------------------------------------
// Repack weight B[256, NOUT] into WMMA fragment order so each lane fetches
// two k-steps of a B fragment with ONE b128 load:
//   Bp4[(h*32 + g)*NOUT + c] = { B[8g+2h][c],   B[8g+2h+1][c],
//                                B[8g+4+2h][c], B[8g+4+2h+1][c] }
// (h = lane>>4, g = k-step pair index, j packed within float4)
// ---------------------------------------------------------------------------
__global__ void repack_b_kernel(const float* __restrict__ B,
                                float* __restrict__ Bp, int NOUT)
{
    const int i = blockIdx.x * blockDim.x + threadIdx.x;   // over 256*NOUT
    if (i >= 256 * NOUT) return;
    const int k = i / NOUT, c = i % NOUT;
    const int h = (k >> 1) & 1;
    const int j = k & 1;
    const int t = k >> 2;
    const int g = t >> 1;
    const int u = t & 1;
    const size_t dst = ((size_t)(h * 32 + g) * NOUT + c) * 4 + u * 2 + j;
    Bp[dst] = B[i];
}

// ---------------------------------------------------------------------------
// fp32 WMMA GEMM: out[M, NOUT] = act(A[M,256] @ B[256,NOUT] (+ bias))
// V_WMMA_F32_16X16X4_F32; one wave owns a 16x64 strip (4 accumulators);
// NOUT/64 waves per block cover a 16-row strip.  B comes pre-packed.
//
// Fragment layouts (ISA 7.12.2, 32-bit matrices, wave32):
//   A 16x4 : lane holds float2 A[lane&15][k0 + 2*(lane>>4) + {0,1}]
//   B 4x16 : lane holds {B[k0+2*(lane>>4)][lane&15], B[k0+2*(lane>>4)+1][..]}
//   C 16x16: vgpr r of lane holds C[r + 8*(lane>>4)][lane&15]
// ---------------------------------------------------------------------------
template<int NOUT, bool RELU, bool BIAS>
__global__ void gemm_kernel(const float* __restrict__ A, int lda,
                            const float4* __restrict__ Bp,    // packed [64, NOUT] float4
                            const float* __restrict__ bias,   // [NOUT] or unused
                            float* __restrict__ out, int ldo,
                            int M)
{
    const int lane = threadIdx.x & 31;
    const int wave = threadIdx.x >> 5;
    const int half = lane >> 4;     // 0: lanes 0-15, 1: lanes 16-31
    const int l15  = lane & 15;
    const int r0   = blockIdx.x * 16;
    const int c0   = wave * 64;
    if (r0 >= M) return;

    v8f acc0 = {}, acc1 = {}, acc2 = {}, acc3 = {};

    const float*  Arow = A + (size_t)(r0 + l15) * lda + 2 * half;
    const float4* Brow = Bp + (size_t)(half * 32) * NOUT + c0 + l15;

    #pragma unroll 4
    for (int g = 0; g < 32; ++g) {          // each g covers k-steps 2g, 2g+1 (K += 8)
        const v2f a0 = *(const v2f*)(Arow + 8 * g);       // k = 8g   + 2h
        const v2f a1 = *(const v2f*)(Arow + 8 * g + 4);   // k = 8g+4 + 2h
        const float4 q0 = Brow[(size_t)g * NOUT + 0];
        const float4 q1 = Brow[(size_t)g * NOUT + 16];
        const float4 q2 = Brow[(size_t)g * NOUT + 32];
        const float4 q3 = Brow[(size_t)g * NOUT + 48];
        acc0 = __builtin_amdgcn_wmma_f32_16x16x4_f32(false, a0, false, (v2f){q0.x, q0.y}, (short)0, acc0, false, false);
        acc1 = __builtin_amdgcn_wmma_f32_16x16x4_f32(false, a0, false, (v2f){q1.x, q1.y}, (short)0, acc1, false, false);
        acc2 = __builtin_amdgcn_wmma_f32_16x16x4_f32(false, a0, false, (v2f){q2.x, q2.y}, (short)0, acc2, false, false);
        acc3 = __builtin_amdgcn_wmma_f32_16x16x4_f32(false, a0, false, (v2f){q3.x, q3.y}, (short)0, acc3, false, false);
        acc0 = __builtin_amdgcn_wmma_f32_16x16x4_f32(false, a1, false, (v2f){q0.z, q0.w}, (short)0, acc0, false, false);
        acc1 = __builtin_amdgcn_wmma_f32_16x16x4_f32(false, a1, false, (v2f){q1.z, q1.w}, (short)0, acc1, false, false);
        acc2 = __builtin_amdgcn_wmma_f32_16x16x4_f32(false, a1, false, (v2f){q2.z, q2.w}, (short)0, acc2, false, false);
        acc3 = __builtin_amdgcn_wmma_f32_16x16x4_f32(false, a1, false, (v2f){q3.z, q3.w}, (short)0, acc3, false, false);
    }

    auto store_tile = [&](v8f c, int cbase) {
        const float bv = BIAS ? bias[cbase + l15] : 0.0f;
        #pragma unroll
        for (int r = 0; r < 8; ++r) {
            const int m = r0 + r + 8 * half;
            float v = c[r] + bv;
            if (RELU) v = fmaxf(v, 0.0f);
            out[(size_t)m * ldo + cbase + l15] = v;
        }
    };
    store_tile(acc0, c0);
    store_tile(acc1, c0 + 16);
    store_tile(acc2, c0 + 32);
    store_tile(acc3, c0 + 48);
}

// ---------------------------------------------------------------------------
// Host-side orchestration (graph-capture safe: async memsets + launches only).
// Workspace: [N,256] fp32 accumulator (51.2 MB) + 4 packed weight buffers.
// ---------------------------------------------------------------------------
extern "C" void kernel_launch(void* const* d_in, const int* in_sizes, int n_in,
                              void* d_out, int out_size, void* d_ws, size_t ws_size,
                              hipStream_t stream) {
    (void)n_in; (void)out_size; (void)ws_size;

    const float* features = (const float*)d_in[0];
    const int*   esrc     = (const int*)  d_in[1];
    const int*   edst     = (const int*)  d_in[2];
    const float* ew       = (const float*)d_in[3];
    const float* W1       = (const float*)d_in[4];
    const float* W2       = (const float*)d_in[5];
    const float* mw1      = (const float*)d_in[6];
    const float* mb1      = (const float*)d_in[7];
    const float* mw2      = (const float*)d_in[8];
    const float* mb2      = (const float*)d_in[9];
    float* out = (float*)d_out;

    const int N = in_sizes[0] / IN_F;   // 50000 (divisible by 16)
    const int E = in_sizes[1];          // 800000

    // ---- workspace carve-up (all 16B aligned) ----
    char* ws = (char*)d_ws;
    float* agg = (float*)ws;                          ws += (size_t)N * HID_F * sizeof(float);
    float* pW1  = (float*)ws;                         ws += (size_t)256 * HID_F * sizeof(float);
    float* pW2  = (float*)ws;                         ws += (size_t)256 * OUT_F * sizeof(float);
    float* pMW1 = (float*)ws;                         ws += (size_t)256 * HID_F * sizeof(float);
    float* pMW2 = (float*)ws;
    const size_t aggBytes = (size_t)N * HID_F * sizeof(float);

    const dim3 spmmGrid(2048), spmmBlock(256);
    const int rowTiles = (N + 15) / 16;

    // pack all four weight matrices into WMMA fragment order
    repack_b_kernel<<<(256 * HID_F) / 256, 256, 0, stream>>>(W1,  pW1,  HID_F);
    repack_b_kernel<<<(256 * OUT_F) / 256, 256, 0, stream>>>(W2,  pW2,  OUT_F);
    repack_b_kernel<<<(256 * HID_F) / 256, 256, 0, stream>>>(mw1, pMW1, HID_F);
    repack_b_kernel<<<(256 * OUT_F) / 256, 256, 0, stream>>>(mw2, pMW2, OUT_F);

    // hop 1: agg = A @ X ; h1 = relu(agg @ W1) -> out[:,128:384]
    hipMemsetAsync(agg, 0, aggBytes, stream);
    spmm_kernel<<<spmmGrid, spmmBlock, 0, stream>>>(features, IN_F, esrc, edst, ew, agg, E);
    gemm_kernel<HID_F, true, false><<<rowTiles, 128, 0, stream>>>(
        agg, HID_F, (const float4*)pW1, nullptr, out + OUT_F, OUT_STRIDE, N);

    // hop 2: agg = A @ h1 (h1 read from out, stride 512); h2 = agg @ W2 -> out[:,384:512]
    hipMemsetAsync(agg, 0, aggBytes, stream);
    spmm_kernel<<<spmmGrid, spmmBlock, 0, stream>>>(out + OUT_F, OUT_STRIDE, esrc, edst, ew, agg, E);
    gemm_kernel<OUT_F, false, false><<<rowTiles, 64, 0, stream>>>(
        agg, HID_F, (const float4*)pW2, nullptr, out + OUT_F + HID_F, OUT_STRIDE, N);

    // self MLP: t = relu(X @ mlp_w1 + b1) (agg reused); self = t @ mlp_w2 + b2 -> out[:,0:128]
    gemm_kernel<HID_F, true, true><<<rowTiles, 128, 0, stream>>>(
        features, IN_F, (const float4*)pMW1, mb1, agg, HID_F, N);
    gemm_kernel<OUT_F, false, true><<<rowTiles, 64, 0, stream>>>(
        agg, HID_F, (const float4*)pMW2, mb2, out, OUT_STRIDE, N);
}